// WholeImageRotationAndTranslation_Undeformable_335007449556
// MI455X (gfx1250) — compile-verified
//
#include <hip/hip_runtime.h>
#include <stdint.h>
#include <math.h>

#define H 128
#define W 128
#define C 64
#define B 16
#define NTHETA 125
#define ROWS_CAP 12           // max fm2 rows one output row can touch (<=7 analytically)
#define CH_GRP 4              // channels staged per async group
#define NGRP (C / CH_GRP)
#define PLANE (H * W)

// ---------------------------------------------------------------------------
// CDNA5 async global->LDS copy of one 16-byte chunk (ASYNCcnt-tracked).
// On gfx1250 a flat pointer to LDS carries the LDS byte offset in its low 32
// bits (ISA 10.2); a flat pointer to global memory
// is numerically the global address.
// ---------------------------------------------------------------------------
__device__ __forceinline__ void async_copy16(const float* g, float* l) {
  asm volatile("global_load_async_to_lds_b128 %0, %1, off"
               :: "v"((unsigned)(uintptr_t)l),
                  "v"((unsigned long long)(uintptr_t)g)
               : "memory");
}
__device__ __forceinline__ void wait_async0() {
  asm volatile("s_wait_asynccnt 0" ::: "memory");
}

__device__ __forceinline__ float waveSum(float v) {
  #pragma unroll
  for (int off = 16; off > 0; off >>= 1) v += __shfl_xor(v, off, 32);
  return v;
}

// ---------------------------------------------------------------------------
// Kernel 1: one block per (theta, b, y). 128 lanes = x coordinates.
// Stages the fm2 row-window needed by this output row per channel group into
// LDS with async copies (double buffered), gathers the 4 bilinear taps from
// LDS, and accumulates sum_c (fm1 - r)^2 weighted by the per-pixel mask.
// ---------------------------------------------------------------------------
__global__ __launch_bounds__(W) void se_kernel(const float* __restrict__ fm1,
                                               const float* __restrict__ fm2,
                                               float* __restrict__ numws,
                                               float* __restrict__ denws) {
  const int x = threadIdx.x;     // 0..127
  const int y = blockIdx.x;      // 0..127
  const int b = blockIdx.y;      // 0..15
  const int t = blockIdx.z;      // 0..124

  // theta (tx outer, ty mid, angle inner); h==w so h/w scale factors are 1.
  // |rr| <= 0.0349 rad, so short Taylor series are exact to f32 precision and
  // avoid the libm sin/cos argument-reduction machinery (scalar-unit only).
  const int   txi = t / 25 - 2;
  const int   tyi = (t / 5) % 5 - 2;
  const int   ai  = t % 5 - 2;
  const float rr  = (float)ai * 0.017453292519943295f;   // pi/180
  const float a2  = rr * rr;
  const float sr  = rr * (1.f + a2 * (-1.f / 6.f + a2 * (1.f / 120.f)));
  const float cr  = 1.f + a2 * (-0.5f + a2 * (1.f / 24.f));
  const float t00 = cr, t01 = -sr, t02 = (float)txi * (1.f / 64.f);
  const float t10 = sr, t11 = cr,  t12 = (float)tyi * (1.f / 64.f);

  // affine_grid (align_corners=True) -> pixel coords
  const float gx = -1.f + (float)x * (2.f / 127.f);
  const float gy = -1.f + (float)y * (2.f / 127.f);
  const float ix = (t00 * gx + t01 * gy + t02 + 1.f) * 0.5f * 127.f;
  const float iy = (t10 * gx + t11 * gy + t12 + 1.f) * 0.5f * 127.f;

  const float x0f = floorf(ix), y0f = floorf(iy);
  const float x1f = x0f + 1.f,  y1f = y0f + 1.f;
  const float wx1 = ix - x0f, wx0 = 1.f - wx1;
  const float wy1 = iy - y0f, wy0 = 1.f - wy1;
  const bool  vx0 = (x0f >= 0.f) & (x0f <= 127.f);
  const bool  vx1 = (x1f >= 0.f) & (x1f <= 127.f);
  const bool  vy0 = (y0f >= 0.f) & (y0f <= 127.f);
  const bool  vy1 = (y1f >= 0.f) & (y1f <= 127.f);
  const float u00 = (vx0 && vy0) ? wx0 * wy0 : 0.f;
  const float u01 = (vx1 && vy0) ? wx1 * wy0 : 0.f;
  const float u10 = (vx0 && vy1) ? wx0 * wy1 : 0.f;
  const float u11 = (vx1 && vy1) ? wx1 * wy1 : 0.f;
  const float mask = u00 + u01 + u10 + u11;

  const int xi0 = min(max((int)x0f, 0), 127);
  const int xi1 = min(max((int)x1f, 0), 127);
  const int yi0 = min(max((int)y0f, 0), 127);
  const int yi1 = min(max((int)y1f, 0), 127);

  // block-wide row window of fm2 touched by this output row
  __shared__ int   sYmin, sYmax;
  __shared__ float sred[8];
  if (threadIdx.x == 0) { sYmin = 127; sYmax = 0; }
  __syncthreads();
  atomicMin(&sYmin, yi0);
  atomicMax(&sYmax, yi1);
  __syncthreads();
  const int ymin  = sYmin;
  int       nrows = sYmax - ymin + 1;
  if (nrows > ROWS_CAP) nrows = ROWS_CAP;      // analytically unreachable guard
  const int ry0 = min(yi0 - ymin, nrows - 1);
  const int ry1 = min(yi1 - ymin, nrows - 1);

  const int rowfl = nrows * W;                 // floats per channel window
  const int o00 = ry0 * W + xi0, o01 = ry0 * W + xi1;
  const int o10 = ry1 * W + xi0, o11 = ry1 * W + xi1;

  __shared__ float sbuf[2][CH_GRP * ROWS_CAP * W];   // 48 KB double buffer

  const float* f2base = fm2 + ((size_t)(b * C) * H + ymin) * W;
  const float* f1p    = fm1 + (((size_t)b * C) * H + y) * W + x;

  auto stage = [&](int cg, float* buf) {
    const int n16 = nrows * (W / 4);           // 16B chunks per channel window
    #pragma unroll
    for (int j = 0; j < CH_GRP; ++j) {
      const float* g = f2base + (size_t)(cg * CH_GRP + j) * PLANE;
      float*       l = buf + j * rowfl;
      for (int i = threadIdx.x; i < n16; i += W)
        async_copy16(g + i * 4, l + i * 4);
    }
  };

  stage(0, sbuf[0]);
  float acc = 0.f;
  for (int cg = 0; cg < NGRP; ++cg) {
    wait_async0();                 // group cg resident in LDS (per-wave count)
    __syncthreads();               // visible block-wide; prev compute done
    if (cg + 1 < NGRP) stage(cg + 1, sbuf[(cg + 1) & 1]);   // overlap next copy
    const float* Bf = sbuf[cg & 1];

    // issue the coalesced fm1 channel loads up front so they overlap the taps
    float f1v[CH_GRP];
    #pragma unroll
    for (int j = 0; j < CH_GRP; ++j)
      f1v[j] = f1p[(size_t)(cg * CH_GRP + j) * PLANE];

    #pragma unroll
    for (int j = 0; j < CH_GRP; ++j) {
      const float* Bc = Bf + j * rowfl;
      const float  r  = u00 * Bc[o00] + u01 * Bc[o01] +
                        u10 * Bc[o10] + u11 * Bc[o11];
      const float  d  = f1v[j] - r;
      acc = fmaf(d, d, acc);
    }
  }

  // block reduction (wave32 shfl + LDS across 4 waves)
  float nv = waveSum(acc * mask);
  float dv = waveSum(mask);
  __syncthreads();
  if ((threadIdx.x & 31) == 0) {
    sred[threadIdx.x >> 5]       = nv;
    sred[4 + (threadIdx.x >> 5)] = dv;
  }
  __syncthreads();
  if (threadIdx.x == 0) {
    numws[((size_t)t * B + b) * H + y] = sred[0] + sred[1] + sred[2] + sred[3];
    if (b == 0)
      denws[t * H + y] = sred[4] + sred[5] + sred[6] + sred[7];
  }
}

// ---------------------------------------------------------------------------
// Kernel 2: per batch, reduce row partials over y for each theta, form
// num / (C * sum(mask)), and take the min over the 125 thetas.
// ---------------------------------------------------------------------------
__global__ __launch_bounds__(128) void min_kernel(const float* __restrict__ numws,
                                                  const float* __restrict__ denws,
                                                  float* __restrict__ out) {
  const int b = blockIdx.x;
  __shared__ float sred[8];
  float best = 3.402823466e38f;
  for (int t = 0; t < NTHETA; ++t) {
    float nv = waveSum(numws[((size_t)t * B + b) * H + threadIdx.x]);
    float dv = waveSum(denws[t * H + threadIdx.x]);
    __syncthreads();
    if ((threadIdx.x & 31) == 0) {
      sred[threadIdx.x >> 5]       = nv;
      sred[4 + (threadIdx.x >> 5)] = dv;
    }
    __syncthreads();
    const float ns = sred[0] + sred[1] + sred[2] + sred[3];
    const float ds = sred[4] + sred[5] + sred[6] + sred[7];
    best = fminf(best, ns / ((float)C * ds));
    __syncthreads();
  }
  if (threadIdx.x == 0) out[b] = best;
}

extern "C" void kernel_launch(void* const* d_in, const int* in_sizes, int n_in,
                              void* d_out, int out_size, void* d_ws, size_t ws_size,
                              hipStream_t stream) {
  (void)in_sizes; (void)n_in; (void)out_size; (void)ws_size;
  const float* fm1 = (const float*)d_in[0];
  const float* fm2 = (const float*)d_in[1];
  float* numws = (float*)d_ws;                           // 125*16*128 floats
  float* denws = numws + (size_t)NTHETA * B * H;         // +125*128 floats (~1.04 MB total)

  dim3 grid(H, B, NTHETA);
  se_kernel<<<grid, W, 0, stream>>>(fm1, fm2, numws, denws);
  min_kernel<<<B, 128, 0, stream>>>(numws, denws, (float*)d_out);
}